// CreateGraphCondensation_23562190586017
// MI455X (gfx1250) — compile-verified
//
#include <hip/hip_runtime.h>
#include <math.h>

typedef float v2f __attribute__((ext_vector_type(2)));
typedef float v8f __attribute__((ext_vector_type(8)));

#define KNN_K 5
#define BIGF 1.0e9f
#define THRS 0.5f

// ---------------- Kernel 1: per-point ||x||^2, packed (||x||^2+pen, pen), per-event max score
__global__ void prep_kernel(const float* __restrict__ score,
                            const float* __restrict__ coords,
                            float* __restrict__ xx, float* __restrict__ aux,
                            float* __restrict__ mrss, int S) {
  __shared__ float red[256];
  int b = blockIdx.x;
  float mx = -1.0e30f;
  for (int t = threadIdx.x; t < S; t += blockDim.x) {
    int i = b * S + t;
    float sc = score[i];
    mx = fmaxf(mx, sc);
    float x = coords[3 * i + 0], y = coords[3 * i + 1], z = coords[3 * i + 2];
    float n2 = x * x + y * y + z * z;
    float pen = (sc > THRS) ? 0.0f : BIGF;   // direction==0 (score>thr) may BE neighbour
    xx[i] = n2;
    aux[2 * i + 0] = n2 + pen;               // folded norm+penalty for candidate side
    aux[2 * i + 1] = pen;                    // clamp floor (exactly BIG for non-candidates)
  }
  red[threadIdx.x] = mx;
  __syncthreads();
  for (int s = blockDim.x / 2; s > 0; s >>= 1) {
    if (threadIdx.x < (unsigned)s) red[threadIdx.x] = fmaxf(red[threadIdx.x], red[threadIdx.x + s]);
    __syncthreads();
  }
  if (threadIdx.x == 0) mrss[b] = red[0];
}

// ---------------- Kernel 2: effective threshold = min(0.98*min_b(max score), 0.5)
__global__ void thr_kernel(const float* __restrict__ mrss, float* __restrict__ thr, int B) {
  if (threadIdx.x == 0) {
    float mn = 1.0e30f;
    for (int b = 0; b < B; ++b) mn = fminf(mn, mrss[b]);
    thr[0] = fminf(mn * 0.98f, THRS);
  }
}

// ---------------- Kernel 3: sel_mask output + per-event counts
__global__ void sel_kernel(const float* __restrict__ score, const float* __restrict__ thr,
                           float* __restrict__ o_sel, int* __restrict__ counts, int S) {
  __shared__ int red[256];
  int b = blockIdx.x;
  float t = thr[0];
  int c = 0;
  for (int k = threadIdx.x; k < S; k += blockDim.x) {
    int i = b * S + k;
    bool s = score[i] >= t;
    o_sel[i] = s ? 1.0f : 0.0f;
    c += s ? 1 : 0;
  }
  red[threadIdx.x] = c;
  __syncthreads();
  for (int s = blockDim.x / 2; s > 0; s >>= 1) {
    if (threadIdx.x < (unsigned)s) red[threadIdx.x] += red[threadIdx.x + s];
    __syncthreads();
  }
  if (threadIdx.x == 0) counts[b] = red[0];
}

// ---------------- Kernel 4: exclusive-prefix row splits (rs_up)
__global__ void rsup_kernel(const int* __restrict__ counts, float* __restrict__ o_rsup, int B) {
  if (threadIdx.x == 0) {
    int acc = 0;
    o_rsup[0] = 0.0f;
    for (int b = 0; b < B; ++b) { acc += counts[b]; o_rsup[b + 1] = (float)acc; }
  }
}

// ---------------- Kernel 5: WMMA-based directional KNN (K=5) per event
// One wave per 16-row query tile; loops over all 16-column candidate tiles.
// Gram tile via V_WMMA_F32_16X16X4_F32 (D=3 padded to K=4).
__global__ void __launch_bounds__(32)
knn_kernel(const float* __restrict__ score,
           const float* __restrict__ coords,
           const float* __restrict__ xx,
           const float* __restrict__ aux,
           float* __restrict__ o_nidx,
           float* __restrict__ o_dist,
           float* __restrict__ o_w,
           int S) {
  __shared__ float sd[16][16][KNN_K];
  __shared__ int   si[16][16][KNN_K];

  const int ntiles  = S >> 4;
  const int wave    = blockIdx.x;
  const int event   = wave / ntiles;
  const int rowtile = wave % ntiles;
  const int lane    = threadIdx.x & 31;
  const int half    = lane >> 4;
  const int lc      = lane & 15;

  const int qbase = event * S + rowtile * 16;

  // A fragment (16x4 f32): lanes 0..15 hold (x,y) of point qbase+lc (K=0,1);
  // lanes 16..31 hold (z, 0) (K=2,3). Unconditional in-bounds loads + cndmask.
  const int pa = qbase + lc;
  float a0 = coords[3 * pa + 2 * half];          // half0: x, half1: z
  float a1 = coords[3 * pa + 1];                 // y (discarded for half1)
  a1 = (half == 0) ? a1 : 0.0f;
  v2f Av = {a0, a1};

  float xxq[8];
#pragma unroll
  for (int r = 0; r < 8; ++r) xxq[r] = xx[qbase + half * 8 + r];

  // Per-lane top-K for each of this lane's 8 rows (lane covers a 16-column stripe).
  float bd[8][KNN_K];
  int   bi[8][KNN_K];
#pragma unroll
  for (int r = 0; r < 8; ++r)
#pragma unroll
    for (int k = 0; k < KNN_K; ++k) { bd[r][k] = BIGF; bi[r][k] = -1; }

  for (int jt = 0; jt < ntiles; ++jt) {
    const int cbase = event * S + jt * 16;
    const int pb = cbase + lc;
    float b0 = coords[3 * pb + 2 * half];
    float b1 = coords[3 * pb + 1];
    b1 = (half == 0) ? b1 : 0.0f;
    v2f Bv = {b0, b1};
    v2f ax = *(const v2f*)(aux + 2 * (size_t)(cbase + lc));  // (xx+pen, pen) one b64 load
    const float xxcp = ax.x;
    const float penc = ax.y;

    v8f Cv = {};
    Cv = __builtin_amdgcn_wmma_f32_16x16x4_f32(false, Av, false, Bv, (short)0, Cv,
                                               false, false);

    const bool sametile = (jt == rowtile);
    const int  cidx = cbase + lc;
#pragma unroll
    for (int r = 0; r < 8; ++r) {
      const int m = half * 8 + r;                       // row in tile for this C VGPR
      // valid: max(d2,0); non-candidate: clamped to exactly penc==BIG
      float dd = fmaxf(fmaf(-2.0f, Cv[r], xxq[r] + xxcp), penc);
      if (sametile && (lc == m)) dd = 2.0f * BIGF;      // exclude self
      if (dd < bd[r][KNN_K - 1]) {                      // capped sorted insert
        float dcur = dd; int icur = cidx;
#pragma unroll
        for (int k = 0; k < KNN_K; ++k) {
          bool sw = dcur < bd[r][k];
          float td = sw ? bd[r][k] : dcur;
          int   ti = sw ? bi[r][k] : icur;
          bd[r][k] = sw ? dcur : bd[r][k];
          bi[r][k] = sw ? icur : bi[r][k];
          dcur = td; icur = ti;
        }
      }
    }
  }

  // Stage per-lane candidates to LDS for the cross-lane (16-way) merge.
#pragma unroll
  for (int r = 0; r < 8; ++r) {
    const int row = half * 8 + r;
#pragma unroll
    for (int k = 0; k < KNN_K; ++k) { sd[row][lc][k] = bd[r][k]; si[row][lc][k] = bi[r][k]; }
  }
  __syncthreads();

  if (lane < 16) {
    const int row = lane;
    float fd[KNN_K]; int fi[KNN_K];
#pragma unroll
    for (int k = 0; k < KNN_K; ++k) { fd[k] = BIGF; fi[k] = -1; }
    for (int l = 0; l < 16; ++l) {
#pragma unroll
      for (int k = 0; k < KNN_K; ++k) {
        float dcur = sd[row][l][k];
        int   icur = si[row][l][k];
        if (dcur < fd[KNN_K - 1]) {
#pragma unroll
          for (int q = 0; q < KNN_K; ++q) {
            bool sw = dcur < fd[q];
            float td = sw ? fd[q] : dcur;
            int   ti = sw ? fi[q] : icur;
            fd[q] = sw ? dcur : fd[q];
            fi[q] = sw ? icur : fi[q];
            dcur = td; icur = ti;
          }
        }
      }
    }
    const int gq = qbase + row;
    const bool isq = (score[gq] <= THRS);               // direction==1: may HAVE neighbours
    float od[KNN_K], oi[KNN_K];
#pragma unroll
    for (int k = 0; k < KNN_K; ++k) {
      bool valid = isq && (fd[k] < BIGF);
      oi[k] = valid ? (float)fi[k] : -1.0f;
      od[k] = valid ? fd[k] : 0.0f;
    }
    // weights = softmax(exp(-dist))
    float e[KNN_K]; float mx = -1.0e30f;
#pragma unroll
    for (int k = 0; k < KNN_K; ++k) { e[k] = __expf(-od[k]); mx = fmaxf(mx, e[k]); }
    float ssum = 0.0f;
#pragma unroll
    for (int k = 0; k < KNN_K; ++k) { e[k] = __expf(e[k] - mx); ssum += e[k]; }
    float inv = 1.0f / ssum;
#pragma unroll
    for (int k = 0; k < KNN_K; ++k) {
      o_nidx[gq * KNN_K + k] = oi[k];
      o_dist[gq * KNN_K + k] = od[k];
      o_w[gq * KNN_K + k]    = e[k] * inv;
    }
  }
}

extern "C" void kernel_launch(void* const* d_in, const int* in_sizes, int n_in,
                              void* d_out, int out_size, void* d_ws, size_t ws_size,
                              hipStream_t stream) {
  (void)n_in; (void)out_size; (void)ws_size;
  const float* score  = (const float*)d_in[0];
  const float* coords = (const float*)d_in[1];
  const int N = in_sizes[0];        // score is [N,1]
  const int B = in_sizes[2] - 1;    // rs is [B+1]
  const int S = N / B;

  float* out    = (float*)d_out;
  float* o_nidx = out;
  float* o_dist = out + (size_t)N * KNN_K;
  float* o_w    = out + 2 * (size_t)N * KNN_K;
  float* o_rsup = out + 3 * (size_t)N * KNN_K;
  float* o_sel  = o_rsup + (B + 1);

  float* xx     = (float*)d_ws;       // [N]
  float* aux    = xx + N;             // [2N] packed (xx+pen, pen)
  float* mrss   = aux + 2 * (size_t)N;// [B]
  float* thr    = mrss + B;           // [1]
  int*   counts = (int*)(thr + 1);    // [B]

  prep_kernel<<<B, 256, 0, stream>>>(score, coords, xx, aux, mrss, S);
  thr_kernel<<<1, 32, 0, stream>>>(mrss, thr, B);
  sel_kernel<<<B, 256, 0, stream>>>(score, thr, o_sel, counts, S);
  rsup_kernel<<<1, 32, 0, stream>>>(counts, o_rsup, B);

  const int ntiles = S / 16;
  knn_kernel<<<B * ntiles, 32, 0, stream>>>(score, coords, xx, aux,
                                            o_nidx, o_dist, o_w, S);
}